// Memory_59038620451412
// MI455X (gfx1250) — compile-verified
//
#include <hip/hip_runtime.h>
#include <hip/hip_bf16.h>
#include <math.h>

#define A_SIZE 64
#define N_MEM  1024
#define RH     4
#define IFW    471
#define BATCH  64
#define OUTSZ  256
#define EPS_F  1e-6f

typedef __attribute__((ext_vector_type(2))) float v2f;
typedef __attribute__((ext_vector_type(4))) float v4f;
typedef __attribute__((ext_vector_type(8))) float v8f;

// ---------------------------------------------------------------------------
// Kernel 1: per-batch write-weight computation.
// One block per batch, 1024 threads (one per memory slot).
// ---------------------------------------------------------------------------
__global__ __launch_bounds__(1024) void dnc_write_kernel(
    const float* __restrict__ iface,
    const float* __restrict__ memory,
    const float* __restrict__ read_w,
    const float* __restrict__ write_wts,
    const float* __restrict__ usage_vec,
    float* __restrict__ write_w_out)
{
    __shared__ float s_val[N_MEM];
    __shared__ int   s_idx[N_MEM];
    __shared__ float s_alloc[N_MEM];
    __shared__ float s_key[A_SIZE];
    __shared__ float s_sc[8]; // [0..3]=free_gate, 4=write_strength, 5=write_gate, 6=alloc_gate, 7=key_norm

    const int b   = blockIdx.x;
    const int tid = threadIdx.x;
    const float* ifr = iface + b * IFW;

    if (tid < A_SIZE) s_key[tid] = ifr[tid];
    if (tid < RH)     s_sc[tid]  = 1.f / (1.f + expf(-ifr[192 + tid]));
    if (tid == 64)    s_sc[4]    = 1.f + log1pf(expf(ifr[200]));   // write_strength
    if (tid == 65)    s_sc[5]    = 1.f / (1.f + expf(-ifr[201]));  // write_gate
    if (tid == 66)    s_sc[6]    = 1.f / (1.f + expf(-ifr[202]));  // alloc_gate
    __syncthreads();
    if (tid == 0) {
        float kn = 0.f;
        for (int a = 0; a < A_SIZE; ++a) kn += s_key[a] * s_key[a];
        s_sc[7] = sqrtf(kn);
    }

    // retention & usage
    const v4f rw4 = *(const v4f*)(read_w + (size_t)b * N_MEM * RH + tid * RH);
    float ret = (1.f - s_sc[0] * rw4.x) * (1.f - s_sc[1] * rw4.y) *
                (1.f - s_sc[2] * rw4.z) * (1.f - s_sc[3] * rw4.w);
    const float u = usage_vec[b * N_MEM + tid];
    const float w = write_wts[b * N_MEM + tid];
    const float usage = (u + w - u * w) * ret;

    s_val[tid] = usage;
    s_idx[tid] = tid;

    // stable ascending bitonic sort of (value, index)
    for (int k = 2; k <= N_MEM; k <<= 1) {
        for (int j = k >> 1; j > 0; j >>= 1) {
            __syncthreads();
            const int ixj = tid ^ j;
            if (ixj > tid) {
                float v1 = s_val[tid], v2 = s_val[ixj];
                int   i1 = s_idx[tid], i2 = s_idx[ixj];
                const bool up = ((tid & k) == 0);
                const bool gt = (v1 > v2) || (v1 == v2 && i1 > i2);
                if (gt == up) {
                    s_val[tid] = v2; s_val[ixj] = v1;
                    s_idx[tid] = i2; s_idx[ixj] = i1;
                }
            }
        }
    }
    __syncthreads();

    const float usort = s_val[tid];
    const int   sidx  = s_idx[tid];

    // inclusive product scan (Hillis-Steele) over sorted usage
    for (int off = 1; off < N_MEM; off <<= 1) {
        const float v  = s_val[tid];
        const float pv = (tid >= off) ? s_val[tid - off] : 1.f;
        __syncthreads();
        s_val[tid] = v * pv;
        __syncthreads();
    }
    const float cum = s_val[tid];
    s_alloc[sidx] = (1.f - usort) * cum;   // scatter via inverse permutation
    __syncthreads();

    // cosine similarity of memory row vs write_key
    const float* mrow = memory + (size_t)b * N_MEM * A_SIZE + (size_t)tid * A_SIZE;
    float dot = 0.f, n2 = 0.f;
    for (int a = 0; a < A_SIZE; a += 4) {
        const v4f m = *(const v4f*)(mrow + a);
        dot += m.x * s_key[a] + m.y * s_key[a + 1] + m.z * s_key[a + 2] + m.w * s_key[a + 3];
        n2  += m.x * m.x + m.y * m.y + m.z * m.z + m.w * m.w;
    }
    const float sim = dot / (sqrtf(n2) * s_sc[7] + EPS_F);
    const float val = sim * s_sc[4];

    // softmax over the 1024 slots (reuse s_val for reductions)
    __syncthreads();
    s_val[tid] = val;
    __syncthreads();
    for (int s = 512; s > 0; s >>= 1) {
        if (tid < s) s_val[tid] = fmaxf(s_val[tid], s_val[tid + s]);
        __syncthreads();
    }
    const float mx = s_val[0];
    __syncthreads();
    const float e = expf(val - mx);
    s_val[tid] = e;
    __syncthreads();
    for (int s = 512; s > 0; s >>= 1) {
        if (tid < s) s_val[tid] += s_val[tid + s];
        __syncthreads();
    }
    const float waddr = e / s_val[0];

    const float ag = s_sc[6];
    write_w_out[b * N_MEM + tid] = s_sc[5] * ((1.f - ag) * waddr + ag * s_alloc[tid]);
}

// ---------------------------------------------------------------------------
// Kernel 2: link-matrix update (bandwidth-dominant: ~537 MB traffic).
// One block per (b,i) row, 256 threads x float4 across j.
// Non-temporal on the 268MB link stream (exceeds 192MB L2).
// ---------------------------------------------------------------------------
__global__ __launch_bounds__(256) void link_update_kernel(
    const float* __restrict__ link,
    const float* __restrict__ prec,
    const float* __restrict__ ww,
    float* __restrict__ out)
{
    const int bi  = blockIdx.x;        // b*1024 + i
    const int b   = bi >> 10;
    const int i   = bi & (N_MEM - 1);
    const int tid = threadIdx.x;
    const int j0  = tid * 4;

    const float  w_i = ww[bi];
    const size_t rowoff = (size_t)bi * N_MEM;

    const v4f wj = *(const v4f*)(ww   + b * N_MEM + j0);
    const v4f pj = *(const v4f*)(prec + b * N_MEM + j0);
    const v4f L  = __builtin_nontemporal_load((const v4f*)(link + rowoff + j0));

    const float a = 1.f - w_i;
    v4f o;
    o.x = (i == j0 + 0) ? 0.f : (a + wj.x) * L.x + w_i * pj.x;
    o.y = (i == j0 + 1) ? 0.f : (a + wj.y) * L.y + w_i * pj.y;
    o.z = (i == j0 + 2) ? 0.f : (a + wj.z) * L.z + w_i * pj.z;
    o.w = (i == j0 + 3) ? 0.f : (a + wj.w) * L.w + w_i * pj.w;

    __builtin_nontemporal_store(o, (v4f*)(out + rowoff + j0));
}

// ---------------------------------------------------------------------------
// Kernel 3: memory_output = interface @ W_out + b_out  via V_WMMA_F32_16X16X4_F32.
// 64x471 @ 471x256, f32 end-to-end. 4x16 tiles of 16x16, one wave each.
// Main loop covers K=0..467 guard-free (471 = 4*117 + 3); tail K-step peeled.
// ---------------------------------------------------------------------------
__global__ __launch_bounds__(128) void gemm_out_kernel(
    const float* __restrict__ iface,
    const float* __restrict__ Wout,
    const float* __restrict__ bout,
    float* __restrict__ out)
{
    const int wave = (blockIdx.x * blockDim.x + threadIdx.x) >> 5; // 0..63
    const int lane = threadIdx.x & 31;
    const int tM   = wave >> 4;   // 0..3
    const int tN   = wave & 15;   // 0..15
    const int half = lane >> 4;
    const int lr   = lane & 15;
    const int row  = tM * 16 + lr;
    const int col  = tN * 16 + lr;

    v8f c = {};
    // main loop: all K indices in range, no masking
    for (int kk = 0; kk < 468; kk += 4) {
        const int ka = kk + 2 * half;
        v2f av, bv;
        av.x = iface[row * IFW + ka];
        av.y = iface[row * IFW + ka + 1];
        bv.x = Wout[(size_t)ka * OUTSZ + col];
        bv.y = Wout[(size_t)(ka + 1) * OUTSZ + col];
        c = __builtin_amdgcn_wmma_f32_16x16x4_f32(
                false, av, false, bv, (short)0, c, false, false);
    }
    // tail: K = 468..470 (3 valid columns); only half=1's second element is OOB
    {
        const int ka = 468 + 2 * half;   // 468 or 470
        v2f av, bv;
        av.x = iface[row * IFW + ka];
        av.y = (ka + 1 < IFW) ? iface[row * IFW + ka + 1] : 0.f;
        bv.x = Wout[(size_t)ka * OUTSZ + col];
        bv.y = (ka + 1 < IFW) ? Wout[(size_t)(ka + 1) * OUTSZ + col] : 0.f;
        c = __builtin_amdgcn_wmma_f32_16x16x4_f32(
                false, av, false, bv, (short)0, c, false, false);
    }

    const int N = tN * 16 + lr;
    const float bias = bout[N];
#pragma unroll
    for (int g = 0; g < 8; ++g) {
        const int M = tM * 16 + g + 8 * half;
        out[M * OUTSZ + N] = c[g] + bias;
    }
}

// ---------------------------------------------------------------------------
extern "C" void kernel_launch(void* const* d_in, const int* in_sizes, int n_in,
                              void* d_out, int out_size, void* d_ws, size_t ws_size,
                              hipStream_t stream)
{
    const float* iface     = (const float*)d_in[0]; // (64, 471)
    const float* memory    = (const float*)d_in[1]; // (64, 1024, 64)
    const float* read_w    = (const float*)d_in[2]; // (64, 1024, 4)
    const float* write_wts = (const float*)d_in[3]; // (64, 1024)
    const float* usage_vec = (const float*)d_in[4]; // (64, 1024)
    const float* prec      = (const float*)d_in[5]; // (64, 1024)
    const float* link      = (const float*)d_in[6]; // (64, 1024, 1024)
    const float* Wout      = (const float*)d_in[7]; // (471, 256)
    const float* bout      = (const float*)d_in[8]; // (256,)

    float* out      = (float*)d_out;
    float* out_mem  = out;                              // (64, 256)
    float* out_ww   = out + BATCH * OUTSZ;              // (64, 1024)
    float* out_link = out + BATCH * OUTSZ + BATCH * N_MEM; // (64, 1024, 1024)

    dnc_write_kernel<<<BATCH, 1024, 0, stream>>>(iface, memory, read_w, write_wts,
                                                 usage_vec, out_ww);
    link_update_kernel<<<BATCH * N_MEM, 256, 0, stream>>>(link, prec, out_ww, out_link);
    gemm_out_kernel<<<16, 128, 0, stream>>>(iface, Wout, bout, out_mem);
}